// GraphConvolution_24747601560251
// MI455X (gfx1250) — compile-verified
//
#include <hip/hip_runtime.h>
#include <hip/hip_bf16.h>

#define IN_F  256
#define OUT_F 128

typedef __attribute__((ext_vector_type(2))) float v2f;
typedef __attribute__((ext_vector_type(8))) float v8f;

// ---------------------------------------------------------------------------
// Kernel 1: out[n, f] = bias[f]   (atomics in kernel 3 accumulate on top)
// ---------------------------------------------------------------------------
__global__ __launch_bounds__(256) void gcn_bias_init(const float* __restrict__ bias,
                                                     float* __restrict__ out,
                                                     int total) {
  int i = (blockIdx.x * 256 + threadIdx.x) * 4;
  if (i < total) {
    float4 b = *(const float4*)&bias[i & (OUT_F - 1)];
    *(float4*)&out[i] = b;
  }
}

// ---------------------------------------------------------------------------
// Kernel 2: support = X @ W via V_WMMA_F32_16X16X4_F32
// Block = 256 threads = 8 waves. Block computes rows [m0, m0+16) x all 128 cols.
// Wave w computes the 16x16 tile at columns [16w, 16w+16).
// W (256x128 f32 = 128 KB) + X tile (16x256 = 16 KB) staged in LDS.
// ---------------------------------------------------------------------------
__global__ __launch_bounds__(256) void gcn_gemm_wmma(const float* __restrict__ X,
                                                     const float* __restrict__ W,
                                                     float* __restrict__ support) {
  __shared__ float sW[IN_F * OUT_F];  // 128 KB
  __shared__ float sX[16 * IN_F];     // 16 KB

  const int tid  = threadIdx.x;
  const int wave = tid >> 5;
  const int lane = tid & 31;
  const int m0   = blockIdx.x * 16;

  // Stage W: 32768 floats, 128 per thread as float4s.
  for (int i = tid * 4; i < IN_F * OUT_F; i += 256 * 4) {
    *(float4*)&sW[i] = *(const float4*)&W[i];
  }
  // Stage X tile: 4096 floats, 16 per thread as float4s.
  for (int i = tid * 4; i < 16 * IN_F; i += 256 * 4) {
    int r = i >> 8;           // i / IN_F
    int c = i & (IN_F - 1);   // i % IN_F
    *(float4*)&sX[i] = *(const float4*)&X[(size_t)(m0 + r) * IN_F + c];
  }
  __syncthreads();

  const int n0    = wave * 16;
  const int lrow  = lane & 15;          // M for A-frag, N for B-frag
  const int khalf = (lane >> 4) * 2;    // lanes 0-15 -> K+0/1, lanes 16-31 -> K+2/3

  v8f acc = {};
  const float* aBase = &sX[lrow * IN_F + khalf];
  const float* bBase = &sW[khalf * OUT_F + n0 + lrow];

  #pragma unroll 4
  for (int k = 0; k < IN_F; k += 4) {
    // A 16x4 f32 fragment: {K, K+1} contiguous -> one 8-byte LDS read.
    v2f a = *(const v2f*)(aBase + k);
    // B 4x16 f32 fragment: rows of W are OUT_F apart.
    v2f b;
    b.x = bBase[(size_t)k * OUT_F];
    b.y = bBase[(size_t)(k + 1) * OUT_F];
    acc = __builtin_amdgcn_wmma_f32_16x16x4_f32(
        /*neg_a=*/false, a, /*neg_b=*/false, b,
        /*c_mod=*/(short)0, acc, /*reuse_a=*/false, /*reuse_b=*/false);
  }

  // D layout: VGPR v -> lanes 0-15: M=v, N=lane; lanes 16-31: M=v+8, N=lane-16.
  const int mbase = m0 + ((lane >> 4) << 3);
  const int col   = n0 + lrow;
  #pragma unroll
  for (int v = 0; v < 8; ++v) {
    support[(size_t)(mbase + v) * OUT_F + col] = acc[v];
  }
}

// ---------------------------------------------------------------------------
// Kernel 3: COO scatter. One wave per edge; lane handles 4 contiguous floats.
// out rows (51 MB) + support (51 MB) both live in the 192 MB L2, so the
// gathers and global_atomic_add_f32 stay on-chip.
// ---------------------------------------------------------------------------
__global__ __launch_bounds__(256) void gcn_spmm_scatter(const float* __restrict__ support,
                                                        const float* __restrict__ adj_val,
                                                        const int* __restrict__ adj_row,
                                                        const int* __restrict__ adj_col,
                                                        float* __restrict__ out,
                                                        int n_edges) {
  int e = blockIdx.x * 8 + (threadIdx.x >> 5);
  if (e >= n_edges) return;
  const int lane = threadIdx.x & 31;

  const int   row = adj_row[e];
  const int   col = adj_col[e];
  const float val = adj_val[e];

  const float4 g = *(const float4*)&support[(size_t)col * OUT_F + lane * 4];
  float* o = &out[(size_t)row * OUT_F + lane * 4];
  atomicAdd(o + 0, g.x * val);
  atomicAdd(o + 1, g.y * val);
  atomicAdd(o + 2, g.z * val);
  atomicAdd(o + 3, g.w * val);
}

// ---------------------------------------------------------------------------
extern "C" void kernel_launch(void* const* d_in, const int* in_sizes, int n_in,
                              void* d_out, int out_size, void* d_ws, size_t ws_size,
                              hipStream_t stream) {
  const float* x       = (const float*)d_in[0];
  const float* weight  = (const float*)d_in[1];
  const float* bias    = (const float*)d_in[2];
  const float* adj_val = (const float*)d_in[3];
  const int*   adj_row = (const int*)d_in[4];
  const int*   adj_col = (const int*)d_in[5];
  float*       out     = (float*)d_out;
  float*       support = (float*)d_ws;   // 100000*128*4 = 51.2 MB scratch

  const int n_nodes = in_sizes[0] / IN_F;  // 100000 (divisible by 16)
  const int n_edges = in_sizes[3];         // 1.6M

  // 1) out = broadcast(bias)
  const int total = n_nodes * OUT_F;
  gcn_bias_init<<<(total / 4 + 255) / 256, 256, 0, stream>>>(bias, out, total);

  // 2) support = X @ W  (WMMA f32)
  gcn_gemm_wmma<<<n_nodes / 16, 256, 0, stream>>>(x, weight, support);

  // 3) out += scatter-add over edges
  gcn_spmm_scatter<<<(n_edges + 7) / 8, 256, 0, stream>>>(support, adj_val, adj_row,
                                                          adj_col, out, n_edges);
}